// SemGraphLearner_66941360275894
// MI455X (gfx1250) — compile-verified
//
#include <hip/hip_runtime.h>
#include <hip/hip_bf16.h>
#include <math.h>

typedef __attribute__((ext_vector_type(2))) float v2f;
typedef __attribute__((ext_vector_type(8))) float v8f;

#define B_  8
#define N_  2048
#define D_  512
#define P_  16
#define H_  32
#define PH_ 512   /* P*H */
#define M_  (B_ * N_)

// ---------------------------------------------------------------------------
// Kernel 1: G[m,k] = relu( sum_d X[m,d] * Wf[k,d] )   (NT gemm, f32 WMMA)
//   X  : [M, D]  row-major
//   Wf : [PH, D] row-major (W viewed as [P*H, D])
//   G  : [M, PH] row-major
// Wave tile: 16 (M) x 64 (PH). Block = 8 waves -> 128 x 64 tile.
// ---------------------------------------------------------------------------
__global__ __launch_bounds__(256) void sg_proj_relu_wmma(
    const float* __restrict__ X,
    const float* __restrict__ Wf,
    float* __restrict__ G)
{
  const int lane = threadIdx.x & 31;
  const int wave = threadIdx.x >> 5;
  const int r = lane & 15;        // M (A) / N (B,C) index within fragment
  const int g = lane >> 4;        // K half selector (A/B), M half selector (C)

  const int m0 = blockIdx.y * 128 + wave * 16;
  const int n0 = blockIdx.x * 64;

  const float* arow = X  + (size_t)(m0 + r) * D_;
  const float* br0  = Wf + (size_t)(n0 +  0 + r) * D_;
  const float* br1  = Wf + (size_t)(n0 + 16 + r) * D_;
  const float* br2  = Wf + (size_t)(n0 + 32 + r) * D_;
  const float* br3  = Wf + (size_t)(n0 + 48 + r) * D_;

  v8f acc0 = {}, acc1 = {}, acc2 = {}, acc3 = {};

#pragma unroll 4
  for (int d0 = 0; d0 < D_; d0 += 4) {
    const int o = d0 + 2 * g;
    v2f a  = *(const v2f*)(arow + o);
    v2f b0 = *(const v2f*)(br0 + o);
    v2f b1 = *(const v2f*)(br1 + o);
    v2f b2 = *(const v2f*)(br2 + o);
    v2f b3 = *(const v2f*)(br3 + o);
    acc0 = __builtin_amdgcn_wmma_f32_16x16x4_f32(false, a, false, b0, (short)0, acc0, false, false);
    acc1 = __builtin_amdgcn_wmma_f32_16x16x4_f32(false, a, false, b1, (short)0, acc1, false, false);
    acc2 = __builtin_amdgcn_wmma_f32_16x16x4_f32(false, a, false, b2, (short)0, acc2, false, false);
    acc3 = __builtin_amdgcn_wmma_f32_16x16x4_f32(false, a, false, b3, (short)0, acc3, false, false);
  }

#pragma unroll
  for (int v = 0; v < 8; ++v) {
    const int row = m0 + v + 8 * g;
    float* o = G + (size_t)row * PH_ + n0 + r;
    o[ 0] = fmaxf(acc0[v], 0.0f);
    o[16] = fmaxf(acc1[v], 0.0f);
    o[32] = fmaxf(acc2[v], 0.0f);
    o[48] = fmaxf(acc3[v], 0.0f);
  }
}

// ---------------------------------------------------------------------------
// Kernel 2: att[b][n][m] = (1/P) * sum_k G_b[n,k] * G_b[m,k]   (NT gemm)
//   G_b = G + b*N*PH,  att logits written to out (f32, [B,N,N])
// Wave tile: 16 (n) x 64 (m). Block = 8 waves -> 128 x 64 tile.
// ---------------------------------------------------------------------------
__global__ __launch_bounds__(256) void sg_gram_wmma(
    const float* __restrict__ G,
    float* __restrict__ out)
{
  const int lane = threadIdx.x & 31;
  const int wave = threadIdx.x >> 5;
  const int r = lane & 15;
  const int g = lane >> 4;

  const int b  = blockIdx.z;
  const int n0 = blockIdx.y * 128 + wave * 16;   // rows of att (n index)
  const int c0 = blockIdx.x * 64;                // cols of att (m index)

  const float* Gb = G + (size_t)b * N_ * PH_;

  const float* arow = Gb + (size_t)(n0 + r) * PH_;
  const float* br0  = Gb + (size_t)(c0 +  0 + r) * PH_;
  const float* br1  = Gb + (size_t)(c0 + 16 + r) * PH_;
  const float* br2  = Gb + (size_t)(c0 + 32 + r) * PH_;
  const float* br3  = Gb + (size_t)(c0 + 48 + r) * PH_;

  v8f acc0 = {}, acc1 = {}, acc2 = {}, acc3 = {};

#pragma unroll 4
  for (int k0 = 0; k0 < PH_; k0 += 4) {
    const int o = k0 + 2 * g;
    v2f a  = *(const v2f*)(arow + o);
    v2f b0 = *(const v2f*)(br0 + o);
    v2f b1 = *(const v2f*)(br1 + o);
    v2f b2 = *(const v2f*)(br2 + o);
    v2f b3 = *(const v2f*)(br3 + o);
    acc0 = __builtin_amdgcn_wmma_f32_16x16x4_f32(false, a, false, b0, (short)0, acc0, false, false);
    acc1 = __builtin_amdgcn_wmma_f32_16x16x4_f32(false, a, false, b1, (short)0, acc1, false, false);
    acc2 = __builtin_amdgcn_wmma_f32_16x16x4_f32(false, a, false, b2, (short)0, acc2, false, false);
    acc3 = __builtin_amdgcn_wmma_f32_16x16x4_f32(false, a, false, b3, (short)0, acc3, false, false);
  }

  const float inv_p = 1.0f / (float)P_;
  float* base = out + (size_t)b * N_ * N_;
#pragma unroll
  for (int v = 0; v < 8; ++v) {
    const int row = n0 + v + 8 * g;
    float* o = base + (size_t)row * N_ + c0 + r;
    o[ 0] = acc0[v] * inv_p;
    o[16] = acc1[v] * inv_p;
    o[32] = acc2[v] * inv_p;
    o[48] = acc3[v] * inv_p;
  }
}

// ---------------------------------------------------------------------------
// Kernel 3: column softmax (over axis=1 == row index n) + mask multiply.
// One thread per column m; reads are coalesced (256 consecutive columns per
// block). Online max/sum pass, then normalize+mask write pass.
// ---------------------------------------------------------------------------
__global__ __launch_bounds__(256) void sg_softmax_cols_mask(
    float* __restrict__ att,              // [B, N, N] logits in, probs out
    const int* __restrict__ ctx_mask)     // [B, N]
{
  const int b   = blockIdx.y;
  const int col = blockIdx.x * 256 + threadIdx.x;

  float* base = att + (size_t)b * N_ * N_;
  const int* mb = ctx_mask + (size_t)b * N_;

  // online softmax over n (stride-N reads per thread, coalesced per block)
  float mx = -INFINITY;
  float sum = 0.0f;
  for (int n = 0; n < N_; ++n) {
    const float x = base[(size_t)n * N_ + col];
    const float nm = fmaxf(mx, x);
    sum = sum * __expf(mx - nm) + __expf(x - nm);
    mx = nm;
  }
  const float inv_sum = 1.0f / sum;
  const float cmask = (float)mb[col];

  for (int n = 0; n < N_; ++n) {
    const size_t idx = (size_t)n * N_ + col;
    const float x = base[idx];
    const float rmask = (float)mb[n];
    base[idx] = __expf(x - mx) * inv_sum * cmask * rmask;
  }
}

// ---------------------------------------------------------------------------
// Launch
// ---------------------------------------------------------------------------
extern "C" void kernel_launch(void* const* d_in, const int* in_sizes, int n_in,
                              void* d_out, int out_size, void* d_ws, size_t ws_size,
                              hipStream_t stream) {
  const float* context = (const float*)d_in[0];   // [B, N, D]
  const float* W       = (const float*)d_in[1];   // [P, H, D] == [PH, D]
  const int*   mask    = (const int*)d_in[2];     // [B, N]
  float* out = (float*)d_out;                     // [B, N, N]
  float* G   = (float*)d_ws;                      // [M, PH] = 32 MiB scratch

  // 1) G = relu(X @ Wf^T) : block tile 128x64, grid (PH/64, M/128)
  dim3 blk(256);
  dim3 grd1(PH_ / 64, M_ / 128);
  hipLaunchKernelGGL(sg_proj_relu_wmma, grd1, blk, 0, stream, context, W, G);

  // 2) att = (G_b @ G_b^T) / P : grid (N/64, N/128, B)
  dim3 grd2(N_ / 64, N_ / 128, B_);
  hipLaunchKernelGGL(sg_gram_wmma, grd2, blk, 0, stream, G, out);

  // 3) softmax over axis=1 (columns of each [N,N] slab) + mask
  dim3 grd3(N_ / 256, B_);
  hipLaunchKernelGGL(sg_softmax_cols_mask, grd3, blk, 0, stream, out, mask);
}